// Beamformer_53077205844753
// MI455X (gfx1250) — compile-verified
//
#include <hip/hip_runtime.h>

#ifndef __has_builtin
#define __has_builtin(x) 0
#endif

// Problem geometry (fixed by setup_inputs)
#define N_TX   4
#define N_EL   128
#define N_AX   2048
#define NZ     256
#define NX     256
#define NPIX   (NZ * NX)

#define PIXPB        128   // pixels staged/processed per block in the main pass
#define THREADS_MAIN 256   // 2 pixel slots x 128 elements (8 wave32 waves)

typedef float v2f __attribute__((ext_vector_type(2)));

// ---- gfx1250 async global->LDS path (compile-probed; falls back cleanly) ----
#if __has_builtin(__builtin_amdgcn_global_load_async_to_lds_b32) && \
    __has_builtin(__builtin_amdgcn_s_wait_asynccnt)
#define BF_ASYNC_LDS 1
typedef __attribute__((address_space(1))) int bf_gint;   // global (AS1)
typedef __attribute__((address_space(3))) int bf_lint;   // LDS    (AS3)
static __device__ __forceinline__ void bf_async_g2l_b32(const void* g, void* l) {
    __builtin_amdgcn_global_load_async_to_lds_b32((bf_gint*)g, (bf_lint*)l, 0, 0);
}
#else
#define BF_ASYNC_LDS 0
#endif

// ---------------------------------------------------------------------------
// Main pass (placed first so the disasm snippet shows its codegen, incl. the
// async global->LDS staging): delay-and-interpolate-and-rotate.
// Block = (tx, 128-pixel chunk); lane -> element, so the float2 stores for one
// pixel are 128 consecutive lanes -> 1 KB contiguous, streamed with TH=NT
// (output is write-once; keep caches for the 8 MB L2-resident RF gathers).
// Per (el,pixel): 1 v_sqrt, 2 b64 gathers from the lane's own trace,
// turns-domain phase with raw v_sin/v_cos (revolution-input, so a single
// floorf replaces libm's Payne-Hanek reduction).
// ---------------------------------------------------------------------------
__global__ __launch_bounds__(THREADS_MAIN)
void bf_das_kernel(const float2* __restrict__ data,   // (4,128,2048) float2
                   const float*  __restrict__ grid,   // (P,3)
                   const float*  __restrict__ probe,  // (128,3)
                   const float*  __restrict__ txdel,  // (4,P) from pass 1
                   const float*  __restrict__ fs_p,
                   const float*  __restrict__ c_p,
                   const float*  __restrict__ fc_p,
                   const float*  __restrict__ fn_p,
                   float2* __restrict__ out)          // (4,P,128) float2
{
    __shared__ float s_px[PIXPB], s_py[PIXPB], s_pz[PIXPB], s_td[PIXPB];

    const int tid = threadIdx.x;
    const int tx  = blockIdx.y;
    const int p0  = blockIdx.x * PIXPB;

#if BF_ASYNC_LDS
    if (tid < PIXPB) {
        const int p = p0 + tid;
        bf_async_g2l_b32(grid + 3 * p + 0, &s_px[tid]);
        bf_async_g2l_b32(grid + 3 * p + 1, &s_py[tid]);
        bf_async_g2l_b32(grid + 3 * p + 2, &s_pz[tid]);
        bf_async_g2l_b32(txdel + tx * NPIX + p, &s_td[tid]);
    }
    __builtin_amdgcn_s_wait_asynccnt(0);
#else
    if (tid < PIXPB) {
        const int p = p0 + tid;
        s_px[tid] = grid[3 * p + 0];
        s_py[tid] = grid[3 * p + 1];
        s_pz[tid] = grid[3 * p + 2];
        s_td[tid] = txdel[tx * NPIX + p];
    }
#endif
    __syncthreads();

    const int el   = tid & (N_EL - 1);
    const int psub = tid >> 7;                 // 0 or 1: which pixel slot

    const float ex = probe[3 * el + 0];
    const float ey = probe[3 * el + 1];
    const float ez = probe[3 * el + 2];
    const float fs = fs_p[0];
    const float c  = c_p[0];
    const float fc = fc_p[0];
    const float fn = fn_p[0];

    const float fs_c = fs / c;                 // samples per meter
    const float kd   = fc / fs;                // phase turns per sample
    const float kz   = 2.0f * fc / c;          // phase turns per meter depth
    const float half_inv_fn = 0.5f / fn;

    const float2* __restrict__ trace = data + (size_t)(tx * N_EL + el) * N_AX;
    float2* __restrict__ outp = out + ((size_t)tx * NPIX + p0) * N_EL + el;

    // Warm the lane's private RF trace into cache (gfx1250 global_prefetch_b8)
    __builtin_prefetch(trace, 0, 0);
    __builtin_prefetch(trace + N_AX / 2, 0, 0);

#pragma unroll 2
    for (int i = psub; i < PIXPB; i += (THREADS_MAIN / N_EL)) {
        const float px = s_px[i];
        const float py = s_py[i];
        const float pz = s_pz[i];
        const float td = s_td[i];

        const float dx = px - ex;
        const float dy = py - ey;
        const float dz = pz - ez;
        const float r  = __builtin_amdgcn_sqrtf(dx * dx + dy * dy + dz * dz);
        const float delay = td + r * fs_c;     // rx + tx delay, in samples

        int d0 = (int)floorf(delay);
        d0 = (d0 < 0) ? 0 : ((d0 > N_AX - 1) ? N_AX - 1 : d0);
        const int d1 = (d0 + 1 > N_AX - 1) ? N_AX - 1 : d0 + 1;
        const float w1 = delay - (float)d0;    // matches reference clipped-index
        const float w0 = (float)d1 - delay;    // weights (OOB -> exact zero)

        const float2 iq0 = trace[d0];
        const float2 iq1 = trace[d1];

        const float apod = (fabsf(dx) <= pz * half_inv_fn) ? 1.0f : 0.0f;
        const float ii = (w0 * iq0.x + w1 * iq1.x) * apod;
        const float qq = (w0 * iq0.y + w1 * iq1.y) * apod;

        // theta / 2pi in "turns"; v_sin/v_cos consume revolutions directly.
        float t = kd * delay - kz * pz;
        t -= floorf(t);                        // range-reduce to [0,1)
        const float ct = __builtin_amdgcn_cosf(t);
        const float st = __builtin_amdgcn_sinf(t);

        v2f o;
        o.x = ii * ct - qq * st;
        o.y = qq * ct + ii * st;
        // TH=NT streaming store: 128 lanes -> 1KB contiguous per pixel
        __builtin_nontemporal_store(o, (v2f*)(outp + (size_t)i * N_EL));
    }
}

// ---------------------------------------------------------------------------
// Pass 1: per-pixel transmit delay (in samples) for each tx:
//   txdel[tx,p] = (min_e(t0[tx,e]*c + |pix_p - el_e|) / c - t_init[tx]) * fs
// 65536 threads x 128-element loop; result = 1 MB in d_ws.
// ---------------------------------------------------------------------------
__global__ __launch_bounds__(256)
void bf_txdel_kernel(const float* __restrict__ grid,     // (P,3)
                     const float* __restrict__ probe,    // (128,3)
                     const float* __restrict__ t0,       // (4,128)
                     const float* __restrict__ t_init,   // (4,)
                     const float* __restrict__ fs_p,
                     const float* __restrict__ c_p,
                     float* __restrict__ txdel)          // (4,P)
{
    __shared__ float s_probe[N_EL * 3];
    __shared__ float s_t0[N_TX * N_EL];
    const int tid = threadIdx.x;
    for (int i = tid; i < N_EL * 3; i += 256)  s_probe[i] = probe[i];
    for (int i = tid; i < N_TX * N_EL; i += 256) s_t0[i] = t0[i];
    __syncthreads();

    const int p = blockIdx.x * 256 + tid;
    const float px = grid[3 * p + 0];
    const float py = grid[3 * p + 1];
    const float pz = grid[3 * p + 2];
    const float c  = c_p[0];
    const float fs = fs_p[0];

    float m0 = 3.4e38f, m1 = 3.4e38f, m2 = 3.4e38f, m3 = 3.4e38f;
#pragma unroll 4
    for (int e = 0; e < N_EL; ++e) {
        const float dx = px - s_probe[3 * e + 0];
        const float dy = py - s_probe[3 * e + 1];
        const float dz = pz - s_probe[3 * e + 2];
        const float d  = __builtin_amdgcn_sqrtf(dx * dx + dy * dy + dz * dz);
        m0 = fminf(m0, s_t0[0 * N_EL + e] * c + d);
        m1 = fminf(m1, s_t0[1 * N_EL + e] * c + d);
        m2 = fminf(m2, s_t0[2 * N_EL + e] * c + d);
        m3 = fminf(m3, s_t0[3 * N_EL + e] * c + d);
    }
    const float inv_c = 1.0f / c;
    txdel[0 * NPIX + p] = (m0 * inv_c - t_init[0]) * fs;
    txdel[1 * NPIX + p] = (m1 * inv_c - t_init[1]) * fs;
    txdel[2 * NPIX + p] = (m2 * inv_c - t_init[2]) * fs;
    txdel[3 * NPIX + p] = (m3 * inv_c - t_init[3]) * fs;
}

// ---------------------------------------------------------------------------
extern "C" void kernel_launch(void* const* d_in, const int* in_sizes, int n_in,
                              void* d_out, int out_size, void* d_ws, size_t ws_size,
                              hipStream_t stream)
{
    // setup_inputs order:
    // 0 data, 1 grid, 2 probe_geometry, 3 initial_times, 4 t0_delays,
    // 5 fs, 6 sound_speed, 7 fc, 8 fdemod (unused by reference math), 9 f_number
    const float* data   = (const float*)d_in[0];
    const float* grid   = (const float*)d_in[1];
    const float* probe  = (const float*)d_in[2];
    const float* t_init = (const float*)d_in[3];
    const float* t0     = (const float*)d_in[4];
    const float* fs     = (const float*)d_in[5];
    const float* c      = (const float*)d_in[6];
    const float* fc     = (const float*)d_in[7];
    const float* fn     = (const float*)d_in[9];
    (void)in_sizes; (void)n_in; (void)out_size; (void)ws_size;

    float* txdel = (float*)d_ws;   // N_TX * NPIX floats = 1 MB scratch

    bf_txdel_kernel<<<NPIX / 256, 256, 0, stream>>>(grid, probe, t0, t_init,
                                                    fs, c, txdel);

    dim3 g(NPIX / PIXPB, N_TX);
    bf_das_kernel<<<g, THREADS_MAIN, 0, stream>>>(
        (const float2*)data, grid, probe, txdel, fs, c, fc, fn, (float2*)d_out);
}